// Atom2Residue_26740466384953
// MI455X (gfx1250) — compile-verified
//
#include <hip/hip_runtime.h>
#include <hip/hip_bf16.h>

typedef __attribute__((ext_vector_type(2))) float v2f;
typedef __attribute__((ext_vector_type(8))) float v8f;

#define NUM_COEF 9
#define ATOM_C   16
#define NODE_C   32
#define FAN_IN   48

// D = A(16x4) * B(4x16) + C(16x16), fp32, wave32 WMMA.
__device__ __forceinline__ v8f wmma_f32_4(v2f a, v2f b, v8f c) {
    return __builtin_amdgcn_wmma_f32_16x16x4_f32(
        /*neg_a=*/false, a, /*neg_b=*/false, b,
        /*c_mod=*/(short)0, c, /*reuse_a=*/false, /*reuse_b=*/false);
}

// One wave handles 16 residues (ca-list entries) x all 9 SH coefficients.
__global__ void Atom2Residue_so3linear_wmma(
    const float* __restrict__ atom_agg,     // [A, 9, 16]
    const float* __restrict__ res_emb,      // [R, 9, 32]
    const float* __restrict__ w,            // [3, 32, 48]
    const float* __restrict__ bias,         // [32]
    const int*   __restrict__ backbone_idx, // [4*Rca]
    const int*   __restrict__ ca_res_idx,   // [Rca]
    float*       __restrict__ out,          // [R, 9, 32]
    int num_ca)
{
    const int wave  = (int)((blockIdx.x * blockDim.x + threadIdx.x) >> 5);
    const int lane  = (int)(threadIdx.x & 31);
    const int jbase = wave * 16;
    if (jbase >= num_ca) return;          // wave-uniform: EXEC stays all-ones

    const int t = lane & 15;              // B-matrix N column / A-matrix M row
    const int h = lane >> 4;              // half select (K = 4*kk + v + 2*h)

    int j = jbase + t;
    if (j >= num_ca) j = num_ca - 1;      // clamp tail -> duplicate benign writes

    const int r    = ca_res_idx[j];               // output residue row
    const int arow = backbone_idx[4 * j + 1];     // CA atom row

    const float* __restrict__ aptr = atom_agg + (long)arow * (NUM_COEF * ATOM_C);
    const float* __restrict__ rptr = res_emb  + (long)r    * (NUM_COEF * NODE_C);

    // Bias chunks matching the C/D layout (o = mtile*16 + v + 8*h).
    const v8f bias0 = *(const v8f*)(bias +       8 * h);
    const v8f bias1 = *(const v8f*)(bias + 16 +  8 * h);

    #pragma unroll
    for (int l = 0; l < 3; ++l) {
        const int cs  = (l == 0) ? 0 : ((l == 1) ? 1 : 4);
        const int cnt = (l == 0) ? 1 : ((l == 1) ? 3 : 5);

        // A-fragments for w[l]: two 16-row M tiles, 12 K-steps of 4.
        // A layout: lane t holds row M=t; VGPR v at half h holds K = 4*kk + v + 2*h.
        const float* __restrict__ wl = w + (long)l * (NODE_C * FAN_IN);
        v2f A0[12], A1[12];
        #pragma unroll
        for (int kk = 0; kk < 12; ++kk) {
            A0[kk] = *(const v2f*)(wl + (long)t        * FAN_IN + 4 * kk + 2 * h);
            A1[kk] = *(const v2f*)(wl + (long)(16 + t) * FAN_IN + 4 * kk + 2 * h);
        }

        #pragma unroll
        for (int c = cs; c < cs + cnt; ++c) {
            // ---- Preload ALL B-fragments for this coefficient first, so the
            // loads issue as one clause and overlap the WMMA chain (graduated
            // s_wait_loadcnt instead of a round-trip per fragment).
            // B layout: lane t holds column N=t; VGPR v at half h holds
            // fuse channel i = 4*kk + v + 2*h.  i<16 -> CA atom, else res_emb.
            v2f B[12];
            #pragma unroll
            for (int kk = 0; kk < 12; ++kk) {
                if (kk < 4) {
                    B[kk] = *(const v2f*)(aptr + c * ATOM_C + 4 * kk + 2 * h);
                } else {
                    B[kk] = *(const v2f*)(rptr + c * NODE_C + (4 * kk - 16) + 2 * h);
                }
            }

            v8f acc0 = {};
            v8f acc1 = {};
            #pragma unroll
            for (int kk = 0; kk < 12; ++kk) {
                acc0 = wmma_f32_4(A0[kk], B[kk], acc0);
                acc1 = wmma_f32_4(A1[kk], B[kk], acc1);
            }
            if (c == 0) {                 // bias only on the l=0 coefficient
                acc0 += bias0;
                acc1 += bias1;
            }
            // Store: lane (h,t) holds out[r, c, o] for o = mtile*16 + 8*h + v,
            // v = 0..7 contiguous -> one 32B vector store per M tile.
            // Output is streamed (written once, never re-read): use
            // non-temporal stores so we don't evict the hot w/atom/res lines.
            float* __restrict__ orow = out + ((long)r * NUM_COEF + c) * NODE_C;
            __builtin_nontemporal_store(acc0, (v8f*)(orow +      8 * h));
            __builtin_nontemporal_store(acc1, (v8f*)(orow + 16 + 8 * h));
        }
    }
}

extern "C" void kernel_launch(void* const* d_in, const int* in_sizes, int n_in,
                              void* d_out, int out_size, void* d_ws, size_t ws_size,
                              hipStream_t stream) {
    (void)n_in; (void)out_size; (void)d_ws; (void)ws_size;
    const float* atom_agg     = (const float*)d_in[0];
    const float* res_emb      = (const float*)d_in[1];
    const float* w            = (const float*)d_in[2];
    const float* b            = (const float*)d_in[3];
    const int*   backbone_idx = (const int*)d_in[4];
    const int*   ca_res_idx   = (const int*)d_in[5];
    float*       out          = (float*)d_out;

    const int num_ca = in_sizes[5];             // entries in ca_res_idx

    // 256 threads = 8 wave32 per block; each wave owns 16 residues.
    const int threads = 256;
    const int res_per_block = (threads / 32) * 16;   // 128
    const int blocks = (num_ca + res_per_block - 1) / res_per_block;

    Atom2Residue_so3linear_wmma<<<blocks, threads, 0, stream>>>(
        atom_agg, res_emb, w, b, backbone_idx, ca_res_idx, out, num_ca);
}